// CubicalModel_ISM_78176994722079
// MI455X (gfx1250) — compile-verified
//
#include <hip/hip_runtime.h>

// CDNA5 / gfx1250. wave32. GEMV via V_WMMA_F32_16X16X4_F32 with B = p replicated
// across all 16 N columns (result invariant to N layout).
// HBM-bound: 206 MB streamed once -> ~8.8 us floor at 23.3 TB/s.

typedef __attribute__((ext_vector_type(2))) float v2f;
typedef __attribute__((ext_vector_type(8))) float v8f;

#define HH 28
#define WW 28
#define NROW 784          // 49 * 16
#define QD 32768
#define KSPLIT 8
#define RBLK 49           // row blocks of 16
#define KCH (QD / KSPLIT) // 4096 K per block
#define NWAVE 8
#define KPW (KCH / NWAVE) // 512 K per wave -> 128 WMMAs
#define UNROLL 8          // 8 WMMAs (32 K) per unrolled block

__global__ __launch_bounds__(256)
void gemv_wmma_kernel(const float* __restrict__ X, const float* __restrict__ Y,
                      const float* __restrict__ p, float* __restrict__ partial) {
  // block -> (mat, row-block, k-chunk)
  int bid = blockIdx.x;
  int mat = bid / (RBLK * KSPLIT);
  int rem = bid % (RBLK * KSPLIT);
  int rb  = rem / KSPLIT;
  int kc  = rem % KSPLIT;

  const float* A = (mat == 0) ? X : Y;

  int tid  = threadIdx.x;
  int lane = tid & 31;
  int wave = tid >> 5;

  int m0   = rb * 16;
  int row  = m0 + (lane & 15);
  int koff = (lane >> 4) << 1;   // lanes 0-15 -> K+0..1, lanes 16-31 -> K+2..3

  const float* rowptr = A + (size_t)row * QD + koff;
  const float* pptr   = p + koff;

  int kbase = kc * KCH + wave * KPW;

  v8f c0 = {};
  v8f c1 = {};
  for (int k0 = kbase; k0 < kbase + KPW; k0 += 4 * UNROLL) {
    // Prefetch four windows (128 K) ahead (clamped inside the row; speculative path).
    int kpf = k0 + 4 * UNROLL * 4;
    if (kpf > QD - 8) kpf = QD - 8;
    __builtin_prefetch(rowptr + kpf, 0, 0);

    v2f a[UNROLL], b[UNROLL];
#pragma unroll
    for (int j = 0; j < UNROLL; ++j) {
      a[j] = *(const v2f*)(rowptr + k0 + 4 * j);  // A fragment: 16x4 f32 tile
      b[j] = *(const v2f*)(pptr   + k0 + 4 * j);  // B fragment: p over all N
    }
#pragma unroll
    for (int j = 0; j < UNROLL; j += 2) {
      // 8 args: (neg_a, A, neg_b, B, c_mod, C, reuse_a, reuse_b)
      c0 = __builtin_amdgcn_wmma_f32_16x16x4_f32(false, a[j], false, b[j],
                                                 (short)0, c0, false, false);
      c1 = __builtin_amdgcn_wmma_f32_16x16x4_f32(false, a[j + 1], false, b[j + 1],
                                                 (short)0, c1, false, false);
    }
  }
  v8f c = c0 + c1;  // fixed-order merge of the two accumulator chains

  // D layout: lanes 0-15 hold M = vgpr (N = lane); lanes 16-31 hold M = 8+vgpr.
  // All N columns identical by construction -> read column N=0 from lanes 0/16.
  __shared__ float red[NWAVE][16];
  if (lane == 0) {
#pragma unroll
    for (int j = 0; j < 8; ++j) red[wave][j] = c[j];
  } else if (lane == 16) {
#pragma unroll
    for (int j = 0; j < 8; ++j) red[wave][8 + j] = c[j];
  }
  __syncthreads();

  // Deterministic fixed-order reduction across the 8 waves of this block.
  if (tid < 16) {
    float s = 0.f;
#pragma unroll
    for (int w = 0; w < NWAVE; ++w) s += red[w][tid];
    partial[((size_t)mat * NROW + (size_t)(m0 + tid)) * KSPLIT + kc] = s;
  }
}

__global__ void gather_kernel(const float* __restrict__ partial,
                              const int* __restrict__ inds1,
                              const int* __restrict__ inds2,
                              float* __restrict__ out) {
  int t = blockIdx.x * blockDim.x + threadIdx.x;
  if (t >= 200) return;
  int mat = t / 100;           // 0 -> dgm1 (X), 1 -> dgm2 (Y)
  int i   = t % 100;
  const int* inds = (mat == 0) ? inds1 : inds2;
  int r  = inds[2 * i];
  int cc = inds[2 * i + 1];
  int row = r * WW + cc;       // flattened (28,28) index
  const float* pp = partial + ((size_t)mat * NROW + (size_t)row) * KSPLIT;
  float s = 0.f;
#pragma unroll
  for (int k = 0; k < KSPLIT; ++k) s += pp[k];  // fixed-order K reduction
  out[t] = s;
}

extern "C" void kernel_launch(void* const* d_in, const int* in_sizes, int n_in,
                              void* d_out, int out_size, void* d_ws, size_t ws_size,
                              hipStream_t stream) {
  const float* X     = (const float*)d_in[0];
  const float* Y     = (const float*)d_in[1];
  const float* p     = (const float*)d_in[2];
  const int*   inds1 = (const int*)d_in[3];
  const int*   inds2 = (const int*)d_in[4];
  float* out     = (float*)d_out;
  float* partial = (float*)d_ws;   // 2 * 784 * 8 floats = 50,176 bytes

  gemv_wmma_kernel<<<2 * RBLK * KSPLIT, 256, 0, stream>>>(X, Y, p, partial);
  gather_kernel<<<1, 256, 0, stream>>>(partial, inds1, inds2, out);
}